// AC_86655260164181
// MI455X (gfx1250) — compile-verified
//
#include <hip/hip_runtime.h>

#define NN 20000
#define NE 160000
#define ET (NE + NN)          // edges + self loops
#define NACT 1024

// ---- workspace offsets (floats) ----
#define OFF_HBUF 0            // N*512
#define OFF_OBUF 10240000     // N*512
#define OFF_XBUF 20480000     // N*128
#define OFF_ESRC 23040000     // N*4
#define OFF_EDST 23120000     // N*4
#define OFF_MB   23200000     // N*4 (uint, ordered-max bits)
#define OFF_Z    23280000     // N*4
#define OFF_W    23360000     // ET*4
#define OFF_BNS  24080000     // 2048 (sum[512], sumsq at +512)
#define OFF_POOL 24082048     // 32
#define OFF_IDX  24082080     // 1024 (int)
#define OFF_A1   24083104     // 1024*128
#define OFF_A2   24214176     // 1024*64

typedef float v2f __attribute__((ext_vector_type(2)));
typedef float v8f __attribute__((ext_vector_type(8)));

__device__ __forceinline__ unsigned ordf(float f) {
  unsigned u = __float_as_uint(f);
  return (u & 0x80000000u) ? ~u : (u | 0x80000000u);
}
__device__ __forceinline__ float unordf(unsigned u) {
  unsigned v = (u & 0x80000000u) ? (u & 0x7FFFFFFFu) : ~u;
  return __uint_as_float(v);
}
__device__ __forceinline__ float hash01(unsigned t, unsigned seed) {
  unsigned s = t * 2654435761u + seed * 0x9E3779B9u + 0x85EBCA6Bu;
  s ^= s >> 16; s *= 0x7FEB352Du; s ^= s >> 15; s *= 0x846CA68Bu; s ^= s >> 16;
  return (float)(s >> 8) * (1.0f / 16777216.0f);
}
__device__ __forceinline__ void edge_sd(const int* ei, int e, int& s, int& d) {
  if (e < NE) { s = ei[e]; d = ei[NE + e]; } else { s = e - NE; d = e - NE; }
}

// ---- extract x[:,1:5] ----
__global__ void k_extract(const float* __restrict__ x, float* __restrict__ xb) {
  int t = blockIdx.x * blockDim.x + threadIdx.x;
  if (t >= NN * 4) return;
  int n = t >> 2, k = t & 3;
  xb[t] = x[n * 6 + 1 + k];
}

// ---- WMMA f32 GEMM: H[n][o] = sum_k X[n][k] * W[o][k] ----
// one wave per 16x16 output tile; A=X (16x4 per step), B[k][o]=W[o][k]
__global__ void k_gemm_wmma(const float* __restrict__ X, const float* __restrict__ W,
                            float* __restrict__ Hout, int nrows, int cin, int coutT) {
  int wavesPerBlock = blockDim.x >> 5;
  int wave = blockIdx.x * wavesPerBlock + (threadIdx.x >> 5);
  int ntn = coutT >> 4;
  int tot = (nrows >> 4) * ntn;
  if (wave >= tot) return;                  // wave-uniform: EXEC stays all-1s
  int mt = wave / ntn, nt = wave % ntn;
  int m0 = mt << 4, n0 = nt << 4;
  int lane = threadIdx.x & 31;
  int half = lane >> 4, lr = lane & 15;
  v8f c;
#pragma unroll
  for (int i = 0; i < 8; ++i) c[i] = 0.0f;
  const float* xrow = X + (size_t)(m0 + lr) * cin + 2 * half;
  const float* wrow = W + (size_t)(n0 + lr) * cin + 2 * half;
  for (int k0 = 0; k0 < cin; k0 += 4) {
    if (k0 + 64 < cin) __builtin_prefetch(xrow + k0 + 64, 0, 0);
    v2f a, b;
    a[0] = xrow[k0]; a[1] = xrow[k0 + 1];
    b[0] = wrow[k0]; b[1] = wrow[k0 + 1];
    c = __builtin_amdgcn_wmma_f32_16x16x4_f32(false, a, false, b, (short)0, c,
                                              false, false);
  }
  float* orow = Hout + (size_t)m0 * coutT + n0 + lr;
#pragma unroll
  for (int p = 0; p < 8; ++p) orow[(size_t)(p + 8 * half) * coutT] = c[p];
}

// ---- per-node attention logits ----
__global__ void k_scores(const float* __restrict__ h, const float* __restrict__ as,
                         const float* __restrict__ ad, float* __restrict__ es,
                         float* __restrict__ ed, int C) {
  int t = blockIdx.x * blockDim.x + threadIdx.x;
  if (t >= NN * 4) return;
  int node = t >> 2, hh = t & 3;
  const float* hp = h + (size_t)node * 4 * C + hh * C;
  const float* pa = as + hh * C;
  const float* pd = ad + hh * C;
  float s1 = 0.f, s2 = 0.f;
  for (int c = 0; c < C; ++c) { float v = hp[c]; s1 += v * pa[c]; s2 += v * pd[c]; }
  es[t] = s1; ed[t] = s2;
}

// ---- edge pass 1: segment max via ordered-uint atomicMax ----
__global__ void k_edge_max(const int* __restrict__ ei, const float* __restrict__ es,
                           const float* __restrict__ ed, unsigned* __restrict__ mb) {
  int t = blockIdx.x * blockDim.x + threadIdx.x;
  if (t >= ET * 4) return;
  int e = t >> 2, hh = t & 3, s, d;
  edge_sd(ei, e, s, d);
  float v = es[s * 4 + hh] + ed[d * 4 + hh];
  v = v > 0.f ? v : 0.2f * v;
  atomicMax(mb + d * 4 + hh, ordf(v));
}

// ---- edge pass 2: w = exp(e - m), z = segment_sum(w) ----
__global__ void k_edge_w(const int* __restrict__ ei, const float* __restrict__ es,
                         const float* __restrict__ ed, const unsigned* __restrict__ mb,
                         float* __restrict__ z, float* __restrict__ wb) {
  int t = blockIdx.x * blockDim.x + threadIdx.x;
  if (t >= ET * 4) return;
  int e = t >> 2, hh = t & 3, s, d;
  edge_sd(ei, e, s, d);
  float v = es[s * 4 + hh] + ed[d * 4 + hh];
  v = v > 0.f ? v : 0.2f * v;
  float w = __expf(v - unordf(mb[d * 4 + hh]));
  wb[t] = w;
  atomicAdd(z + d * 4 + hh, w);
}

// ---- edge pass 3: out[dst] += h[src] * alpha (float4 chunks) ----
__global__ void k_edge_agg(const int* __restrict__ ei, const float* __restrict__ hb,
                           const float* __restrict__ wb, const float* __restrict__ z,
                           float* __restrict__ ob, int C) {
  long long t = (long long)blockIdx.x * blockDim.x + threadIdx.x;
  int chunks = C;  // HC/4 = 4C/4
  if (t >= (long long)ET * chunks) return;
  int e = (int)(t / chunks);
  int j4 = (int)(t % chunks) * 4;
  int hh = j4 / C;
  int s, d;
  edge_sd(ei, e, s, d);
  float alpha = wb[e * 4 + hh] / (z[d * 4 + hh] + 1e-16f);
  float4 hv = *(const float4*)(hb + (size_t)s * 4 * C + j4);
  float* op = ob + (size_t)d * 4 * C + j4;
  atomicAdd(op + 0, hv.x * alpha);
  atomicAdd(op + 1, hv.y * alpha);
  atomicAdd(op + 2, hv.z * alpha);
  atomicAdd(op + 3, hv.w * alpha);
}

// ---- mean over heads + bias -> xbuf ----
__global__ void k_reduce_heads(const float* __restrict__ ob, const float* __restrict__ bias,
                               float* __restrict__ xb, int C) {
  int t = blockIdx.x * blockDim.x + threadIdx.x;
  if (t >= NN * C) return;
  int node = t / C, c = t % C;
  const float* p = ob + (size_t)node * 4 * C + c;
  xb[t] = (p[0] + p[C] + p[2 * C] + p[3 * C]) * 0.25f + bias[c];
}

// ---- per-channel sum/sumsq for BN ----
__global__ void k_colstats(const float* __restrict__ xb, float* __restrict__ bns, int C) {
  __shared__ float s1[256], s2[256];
  int c = blockIdx.x, t = threadIdx.x;
  float s = 0.f, q = 0.f;
  for (int i = t; i < NN; i += 256) {
    float v = xb[(size_t)i * C + c];
    s += v; q += v * v;
  }
  s1[t] = s; s2[t] = q; __syncthreads();
  for (int st = 128; st > 0; st >>= 1) {
    if (t < st) { s1[t] += s1[t + st]; s2[t] += s2[t + st]; }
    __syncthreads();
  }
  if (t == 0) { bns[c] = s1[0]; bns[512 + c] = s2[0]; }
}

// ---- BN + (ELU) + dropout in-place ----
__global__ void k_bn_apply(float* __restrict__ xb, const float* __restrict__ bns,
                           const float* __restrict__ g, const float* __restrict__ b,
                           int C, int do_elu, unsigned seed) {
  int t = blockIdx.x * blockDim.x + threadIdx.x;
  if (t >= NN * C) return;
  int c = t % C;
  float mu = bns[c] * (1.0f / NN);
  float var = bns[512 + c] * (1.0f / NN) - mu * mu;
  float y = (xb[t] - mu) * rsqrtf(var + 1e-5f) * g[c] + b[c];
  if (do_elu) y = y > 0.f ? y : (__expf(y) - 1.f);
  y = (hash01((unsigned)t, seed) >= 0.5f) ? y * 2.f : 0.f;
  xb[t] = y;
}

// ---- mean pool over nodes ----
__global__ void k_pool(const float* __restrict__ xb, float* __restrict__ pool, int C) {
  __shared__ float s1[256];
  int c = blockIdx.x, t = threadIdx.x;
  float s = 0.f;
  for (int i = t; i < NN; i += 256) s += xb[(size_t)i * C + c];
  s1[t] = s; __syncthreads();
  for (int st = 128; st > 0; st >>= 1) {
    if (t < st) s1[t] += s1[t + st];
    __syncthreads();
  }
  if (t == 0) pool[c] = s1[0] * (1.0f / NN);
}

// ---- ordered compaction of action nodes (single wave32 + ballot) ----
__global__ void k_compact(const float* __restrict__ x, int* __restrict__ idx) {
  int lane = threadIdx.x;
  int cnt = 0;
  for (int base = 0; base < NN; base += 32) {
    int i = base + lane;
    bool c = false;
    if (i < NN) c = (x[i * 6 + 2] == 1.0f) && (x[i * 6 + 5] == 0.0f);
    unsigned long long m = __ballot((int)c);
    int pos = __popcll(m & ((1ull << lane) - 1ull));
    if (c) { int w = cnt + pos; if (w < NACT) idx[w] = i; }
    cnt += __popcll(m);
  }
}

// ---- MLP layer: out = relu(LN(in[row] @ W^T + b)) ; block per row ----
__global__ void k_mlp(const float* __restrict__ in, const int* __restrict__ idx,
                      const float* __restrict__ w, const float* __restrict__ b,
                      const float* __restrict__ g, const float* __restrict__ beta,
                      float* __restrict__ out, int din, int dout) {
  __shared__ float s1[128], s2[128];
  int r = blockIdx.x, j = threadIdx.x;
  const float* row = in + (size_t)(idx ? idx[r] : r) * din;
  float acc = b[j];
  const float* wr = w + (size_t)j * din;
  for (int k = 0; k < din; ++k) acc += row[k] * wr[k];
  s1[j] = acc; s2[j] = acc * acc; __syncthreads();
  for (int st = dout >> 1; st > 0; st >>= 1) {
    if (j < st) { s1[j] += s1[j + st]; s2[j] += s2[j + st]; }
    __syncthreads();
  }
  float mu = s1[0] / (float)dout;
  float var = s2[0] / (float)dout - mu * mu;
  float y = (acc - mu) * rsqrtf(var + 1e-5f) * g[j] + beta[j];
  out[(size_t)r * dout + j] = fmaxf(y, 0.f);
}

// ---- actor head: tanh(w3 . a2 + b3), softmax + log_softmax -> d_out ----
__global__ void k_actor_final(const float* __restrict__ a2, const float* __restrict__ w3,
                              const float* __restrict__ b3, float* __restrict__ dout,
                              int din) {
  __shared__ float sm[1024];
  int t = threadIdx.x;
  float acc = b3[0];
  const float* row = a2 + (size_t)t * din;
  for (int k = 0; k < din; ++k) acc += row[k] * w3[k];
  float logit = tanhf(acc);
  sm[t] = logit; __syncthreads();
  for (int st = 512; st > 0; st >>= 1) {
    if (t < st) sm[t] = fmaxf(sm[t], sm[t + st]);
    __syncthreads();
  }
  float mx = sm[0]; __syncthreads();
  float ex = __expf(logit - mx);
  sm[t] = ex; __syncthreads();
  for (int st = 512; st > 0; st >>= 1) {
    if (t < st) sm[t] += sm[t + st];
    __syncthreads();
  }
  float Z = sm[0];
  dout[t] = ex / Z;                                  // prob
  dout[1025 + t] = (logit - mx) - __logf(Z);         // log_prob
}

// ---- critic head: single block ----
__global__ void k_critic(const float* __restrict__ pooled,
                         const float* __restrict__ w1, const float* __restrict__ b1,
                         const float* __restrict__ g1, const float* __restrict__ be1,
                         const float* __restrict__ w2, const float* __restrict__ b2,
                         const float* __restrict__ g2, const float* __restrict__ be2,
                         const float* __restrict__ w3, const float* __restrict__ b3,
                         float* __restrict__ dout) {
  __shared__ float c1[128], c2[32], red[2], s1[128], s2[128];
  int j = threadIdx.x;
  float acc = b1[j];
  for (int k = 0; k < 32; ++k) acc += pooled[k] * w1[j * 32 + k];
  s1[j] = acc; s2[j] = acc * acc; __syncthreads();
  for (int st = 64; st > 0; st >>= 1) {
    if (j < st) { s1[j] += s1[j + st]; s2[j] += s2[j + st]; }
    __syncthreads();
  }
  float mu = s1[0] / 128.f, var = s2[0] / 128.f - (s1[0] / 128.f) * (s1[0] / 128.f);
  c1[j] = fmaxf((acc - mu) * rsqrtf(var + 1e-5f) * g1[j] + be1[j], 0.f);
  __syncthreads();
  float acc2 = 0.f;
  if (j < 32) {
    acc2 = b2[j];
    for (int k = 0; k < 128; ++k) acc2 += c1[k] * w2[j * 128 + k];
    c2[j] = acc2;
  }
  __syncthreads();
  if (j == 0) {
    float s = 0.f, q = 0.f;
    for (int k = 0; k < 32; ++k) { s += c2[k]; q += c2[k] * c2[k]; }
    float m = s / 32.f;
    red[0] = m; red[1] = q / 32.f - m * m;
  }
  __syncthreads();
  if (j < 32) c2[j] = fmaxf((acc2 - red[0]) * rsqrtf(red[1] + 1e-5f) * g2[j] + be2[j], 0.f);
  __syncthreads();
  if (j == 0) {
    float v = b3[0];
    for (int k = 0; k < 32; ++k) v += c2[k] * w3[k];
    dout[1024] = v;
  }
}

extern "C" void kernel_launch(void* const* d_in, const int* in_sizes, int n_in,
                              void* d_out, int out_size, void* d_ws, size_t ws_size,
                              hipStream_t stream) {
  (void)in_sizes; (void)n_in; (void)out_size; (void)ws_size;
  const float* x = (const float*)d_in[0];
  const int* ei = (const int*)d_in[1];
  const float* gw[3]  = {(const float*)d_in[3], (const float*)d_in[9],  (const float*)d_in[15]};
  const float* gas[3] = {(const float*)d_in[4], (const float*)d_in[10], (const float*)d_in[16]};
  const float* gad[3] = {(const float*)d_in[5], (const float*)d_in[11], (const float*)d_in[17]};
  const float* gb[3]  = {(const float*)d_in[6], (const float*)d_in[12], (const float*)d_in[18]};
  const float* bng[3] = {(const float*)d_in[7], (const float*)d_in[13], (const float*)d_in[19]};
  const float* bnb[3] = {(const float*)d_in[8], (const float*)d_in[14], (const float*)d_in[20]};
  const int cins[3] = {4, 128, 64};
  const int Cs[3]   = {128, 64, 32};

  float* ws = (float*)d_ws;
  float* hbuf = ws + OFF_HBUF;
  float* obuf = ws + OFF_OBUF;
  float* xbuf = ws + OFF_XBUF;
  float* esrc = ws + OFF_ESRC;
  float* edst = ws + OFF_EDST;
  unsigned* mbits = (unsigned*)(ws + OFF_MB);
  float* zbuf = ws + OFF_Z;
  float* wbuf = ws + OFF_W;
  float* bns  = ws + OFF_BNS;
  float* pool = ws + OFF_POOL;
  int* idx    = (int*)(ws + OFF_IDX);
  float* a1   = ws + OFF_A1;
  float* a2   = ws + OFF_A2;

  k_extract<<<(NN * 4 + 255) / 256, 256, 0, stream>>>(x, xbuf);

  for (int li = 0; li < 3; ++li) {
    int cin = cins[li], C = Cs[li], HC = 4 * C;
    int tiles = (NN / 16) * (HC / 16);
    k_gemm_wmma<<<(tiles + 7) / 8, 256, 0, stream>>>(xbuf, gw[li], hbuf, NN, cin, HC);
    k_scores<<<(NN * 4 + 255) / 256, 256, 0, stream>>>(hbuf, gas[li], gad[li], esrc, edst, C);
    hipMemsetAsync(mbits, 0, (size_t)NN * 4 * sizeof(unsigned), stream);
    hipMemsetAsync(zbuf, 0, (size_t)NN * 4 * sizeof(float), stream);
    hipMemsetAsync(obuf, 0, (size_t)NN * HC * sizeof(float), stream);
    k_edge_max<<<(ET * 4 + 255) / 256, 256, 0, stream>>>(ei, esrc, edst, mbits);
    k_edge_w<<<(ET * 4 + 255) / 256, 256, 0, stream>>>(ei, esrc, edst, mbits, zbuf, wbuf);
    long long aggN = (long long)ET * C;
    k_edge_agg<<<(unsigned)((aggN + 255) / 256), 256, 0, stream>>>(ei, hbuf, wbuf, zbuf, obuf, C);
    k_reduce_heads<<<(NN * C + 255) / 256, 256, 0, stream>>>(obuf, gb[li], xbuf, C);
    k_colstats<<<C, 256, 0, stream>>>(xbuf, bns, C);
    k_bn_apply<<<(NN * C + 255) / 256, 256, 0, stream>>>(xbuf, bns, bng[li], bnb[li], C,
                                                         (li < 2) ? 1 : 0,
                                                         0x1234u + (unsigned)li * 0x1111u);
  }

  k_pool<<<32, 256, 0, stream>>>(xbuf, pool, 32);
  k_compact<<<1, 32, 0, stream>>>(x, idx);

  // actor: 32 -> 128 -> 64 -> 1(tanh) over 1024 selected rows
  k_mlp<<<NACT, 128, 0, stream>>>(xbuf, idx, (const float*)d_in[21], (const float*)d_in[22],
                                  (const float*)d_in[23], (const float*)d_in[24], a1, 32, 128);
  k_mlp<<<NACT, 64, 0, stream>>>(a1, nullptr, (const float*)d_in[25], (const float*)d_in[26],
                                 (const float*)d_in[27], (const float*)d_in[28], a2, 128, 64);
  k_actor_final<<<1, 1024, 0, stream>>>(a2, (const float*)d_in[29], (const float*)d_in[30],
                                        (float*)d_out, 64);
  // critic on pooled
  k_critic<<<1, 128, 0, stream>>>(pool,
                                  (const float*)d_in[31], (const float*)d_in[32],
                                  (const float*)d_in[33], (const float*)d_in[34],
                                  (const float*)d_in[35], (const float*)d_in[36],
                                  (const float*)d_in[37], (const float*)d_in[38],
                                  (const float*)d_in[39], (const float*)d_in[40],
                                  (float*)d_out);
}